// CTCLayer_36670430773381
// MI455X (gfx1250) — compile-verified
//
#include <hip/hip_runtime.h>
#include <cstdint>

// CTC batch cost, forward (alpha) recursion in log space — wave-per-batch.
//
// Grid: 512 workgroups of ONE wave32 each (one batch per wave, no barriers).
// Lane i owns extended states 4i..4i+3 in registers; cross-lane neighbor
// alphas move via __shfl_up (wave-synchronous). Probability rows y_pred[b,t,:]
// (512 B) stream into LDS through CDNA5 async global->LDS copies: one
// global_load_async_to_lds_b128 per row (32 lanes x 16 B), pipeline depth 32
// (ASYNCcnt is 6-bit), i.e. ~8 MB in flight chip-wide against the
// 23.3 TB/s * ~1us bandwidth-latency product. HBM floor: 134 MB -> ~5.8 us.

#define B_    512
#define T_    512
#define C_    128
#define L_    48
#define S_    (2 * L_ + 1)   // 97
#define BLANK (C_ - 1)
#define NEGV  (-1e30f)
#define EPSV  (1e-7f)
#define DEPTH 32             // rows in flight per wave (<= 63, ASYNCcnt is 6-bit)

#if defined(__has_builtin)
#if __has_builtin(__builtin_amdgcn_s_wait_asynccnt)
#define WAIT_ASYNC(n) __builtin_amdgcn_s_wait_asynccnt(n)
#endif
#endif
#ifndef WAIT_ASYNC
#define WAIT_ASYNC(n) asm volatile("s_wait_asynccnt " #n ::: "memory")
#endif

// One whole 512-B row per instruction: each lane copies 16 B global->LDS.
// GVS mode: mem = SGPR64 base + VGPR32 byte offset; LDS dest address in VGPR.
__device__ __forceinline__ void async_copy_b128(uint32_t lds_addr, uint32_t goff,
                                                uint64_t sbase) {
  asm volatile("global_load_async_to_lds_b128 %0, %1, %2"
               :: "v"(lds_addr), "v"(goff), "s"(sbase)
               : "memory");
}

// Fused 3-way logaddexp: max3 + 3x v_exp_f32 + 1x v_log_f32.
__device__ __forceinline__ float lae3(float a, float b, float c) {
  float m = fmaxf(fmaxf(a, b), c);
  return m + __logf(__expf(a - m) + __expf(b - m) + __expf(c - m));
}

__device__ __forceinline__ float lae2(float a, float b) {
  float m = fmaxf(a, b);
  return m + __logf(__expf(a - m) + __expf(b - m));
}

__global__ __launch_bounds__(32)
void ctc_alpha_wave(const int* __restrict__ y_true,
                    const float* __restrict__ y_pred,
                    float* __restrict__ out) {
  __shared__ float row[DEPTH][C_];   // 32 x 512 B = 16 KB pipelined rows

  const int lane = threadIdx.x;      // wave32: 0..31
  const int b    = blockIdx.x;

  // ---- per-lane extended symbols for states s = 4*lane + e ----
  int  ext1 = BLANK, ext3 = BLANK;   // odd states (labels); even states = blank
  bool sk0 = false, sk1 = false, sk2 = false, sk3 = false;
  {
    const int s1 = 4 * lane + 1;
    const int s3 = 4 * lane + 3;
    if (s1 < S_) {                   // odd state -> label index s1>>1
      const int li = s1 >> 1;
      ext1 = y_true[b * L_ + li];
      sk1  = (li == 0) || (ext1 != y_true[b * L_ + li - 1]);
    }
    if (s3 < S_) {
      const int li = s3 >> 1;
      ext3 = y_true[b * L_ + li];
      sk3  = (ext3 != y_true[b * L_ + li - 1]);   // li >= 1 always here
    }
    // even states never skip (ext == blank)
  }

  const uint64_t base    = (uint64_t)(uintptr_t)(y_pred + (size_t)b * T_ * C_);
  const uint32_t rowbase = (uint32_t)(uintptr_t)(&row[0][0]);
  const uint32_t laneoff = (uint32_t)(lane * 16);   // 16 B per lane per row

  // ---- prologue: fill async pipeline with rows 0..DEPTH-1 ----
#pragma unroll
  for (int k = 0; k < DEPTH; ++k) {
    async_copy_b128(rowbase + (uint32_t)(k * C_ * 4) + laneoff,
                    (uint32_t)(k * C_ * 4) + laneoff, base);
  }

  float a0 = NEGV, a1 = NEGV, a2 = NEGV, a3 = NEGV;

  for (int t = 0; t < T_; ++t) {
    WAIT_ASYNC(31);                 // oldest outstanding copy (row t) landed
    const float* r = row[t & (DEPTH - 1)];

    // gathered log-probs for this lane's 4 states (even states share blank)
    const float lpb = __logf(r[BLANK] + EPSV);       // LDS broadcast read
    const float lp1 = __logf(r[ext1] + EPSV);
    const float lp3 = __logf(r[ext3] + EPSV);

    // neighbor alphas from lane-1 (old values), wave-synchronous
    float p3 = __shfl_up(a3, 1, 32);  // alpha[4*lane - 1]
    float p2 = __shfl_up(a2, 1, 32);  // alpha[4*lane - 2]
    if (lane == 0) { p3 = NEGV; p2 = NEGV; }

    if (t == 0) {
      // alpha0: only states 0 (blank) and 1 (first label) are live
      a0 = (lane == 0) ? lpb : NEGV;
      a1 = (lane == 0) ? lp1 : NEGV;
      a2 = NEGV;
      a3 = NEGV;
    } else {
      const float n0 = lae3(a0, p3, sk0 ? p2 : NEGV) + lpb;
      const float n1 = lae3(a1, a0, sk1 ? p3 : NEGV) + lp1;
      const float n2 = lae3(a2, a1, sk2 ? a0 : NEGV) + lpb;
      const float n3 = lae3(a3, a2, sk3 ? a1 : NEGV) + lp3;
      a0 = n0; a1 = n1; a2 = n2; a3 = n3;
    }

    // row-t LDS reads must retire before its slot is overwritten
    asm volatile("s_wait_dscnt 0" ::: "memory");

    const int tn = t + DEPTH;
    if (tn < T_) {
      async_copy_b128(rowbase + (uint32_t)((t & (DEPTH - 1)) * C_ * 4) + laneoff,
                      (uint32_t)(tn * C_ * 4) + laneoff, base);
    }
  }

  // ---- loss = -logaddexp(alpha[S-1], alpha[S-2]) = states 96, 95 ----
  // state 96 = lane 24 elem 0; state 95 = lane 23 elem 3
  const float v96 = __shfl(a0, 24, 32);
  const float v95 = __shfl(a3, 23, 32);
  if (lane == 0) {
    out[b] = -lae2(v96, v95);
  }
}

extern "C" void kernel_launch(void* const* d_in, const int* in_sizes, int n_in,
                              void* d_out, int out_size, void* d_ws, size_t ws_size,
                              hipStream_t stream) {
  const int*   y_true = (const int*)d_in[0];    // [B, L] int32
  const float* y_pred = (const float*)d_in[1];  // [B, T, C] float32 probs
  float*       out    = (float*)d_out;          // [B, 1] float32
  (void)in_sizes; (void)n_in; (void)out_size; (void)d_ws; (void)ws_size;

  ctc_alpha_wave<<<B_, 32, 0, stream>>>(y_true, y_pred, out);
}